// ProjectionSimpleNetPyBatched_58729382806282
// MI455X (gfx1250) — compile-verified
//
#include <hip/hip_runtime.h>
#include <cstdint>

#define D 64
#define LDSP 65                 // padded LDS row stride in floats (64 banks -> conflict-free columns)
#define NTHR 128                // 4 wave32
#define N_BISECT 40
#define ETA_MAX 10000.0f
#define LOG2PIE 2.837877066409345f   // log(2*pi) + 1

typedef float v2f  __attribute__((ext_vector_type(2)));
typedef float v8f  __attribute__((ext_vector_type(8)));
typedef unsigned int u32x4 __attribute__((ext_vector_type(4)));
typedef int   i32x4 __attribute__((ext_vector_type(4)));
typedef int   i32x8 __attribute__((ext_vector_type(8)));

// ---------------------------------------------------------------------------
// TDM: stage a 64x64 fp32 tile (row-major, contiguous in global) into LDS with
// a 65-float padded row stride. D# per CDNA5 ISA:
//   group0: count=1, lds_addr, global_addr[56:0], type=2
//   group1: data_size=4B, pad_enable, pad_interval=64 dwords, pad_amount=1 dword,
//           tensor_dim0=64, tensor_dim1=64, tile_dim0=64, tile_dim1=64,
//           tensor_dim0_stride=64
// Called by one wave; completion via s_wait_tensorcnt.
// ---------------------------------------------------------------------------
__device__ inline void tdm_load_64x64(float* lds_dst, const float* gsrc) {
#if __has_builtin(__builtin_amdgcn_tensor_load_to_lds)
  const uint64_t ga = (uint64_t)(uintptr_t)gsrc;
  const uint32_t la = (uint32_t)(uintptr_t)lds_dst;   // low 32 bits = LDS byte offset
  u32x4 g0;
  g0[0] = 1u;                                          // count=1, user descriptor
  g0[1] = la;                                          // lds_addr
  g0[2] = (uint32_t)ga;                                // global_addr[31:0]
  g0[3] = ((uint32_t)(ga >> 32) & 0x1FFFFFFu) | (2u << 30);  // addr[56:32] | type=2
  i32x8 g1;
  g1[0] = (2 << 16) | (1 << 20) | (5 << 22);           // data_size=4B | pad_enable | pad_interval=64dw, pad_amount=1dw
  g1[1] = (int)(64u << 16);                            // tensor_dim0 = 64 (low 16)
  g1[2] = (int)(64u << 16);                            // tensor_dim0 hi=0 | tensor_dim1 = 64 (low 16)
  g1[3] = (int)(64u << 16);                            // tensor_dim1 hi=0 | tile_dim0 = 64
  g1[4] = 64;                                          // tile_dim1 = 64, tile_dim2 = 0
  g1[5] = 64;                                          // tensor_dim0_stride = 64 (low 32)
  g1[6] = 0;
  g1[7] = 0;
  i32x4 z4 = {0, 0, 0, 0};
#if defined(__clang_major__) && (__clang_major__ >= 23)
  i32x8 z8 = {0, 0, 0, 0, 0, 0, 0, 0};
  __builtin_amdgcn_tensor_load_to_lds(g0, g1, z4, z4, z8, 0);
#else
  __builtin_amdgcn_tensor_load_to_lds(g0, g1, z4, z4, 0);
#endif
#else
  // Fallback: cooperative copy by the calling wave
  const int lane = threadIdx.x & 31;
  for (int idx = lane; idx < D * D; idx += 32) {
    const int i = idx >> 6, j = idx & 63;
    lds_dst[i * LDSP + j] = gsrc[idx];
  }
#endif
}

// Block-wide sum via ds_add_f32 on an LDS accumulator. Safe for back-to-back use.
__device__ inline float block_sum(float local, float* red, int tid) {
  if (tid == 0) *red = 0.0f;
  __syncthreads();
  atomicAdd(red, local);
  __syncthreads();
  const float r = *red;
  __syncthreads();
  return r;
}

// In-place Cholesky (lower) of a 64x64 SPD matrix in padded LDS.
__device__ inline void chol_inplace(float* A, int tid) {
  for (int k = 0; k < D; ++k) {
    if (tid == 0) A[k * LDSP + k] = sqrtf(A[k * LDSP + k]);
    __syncthreads();
    const float dkk = A[k * LDSP + k];
    for (int i = k + 1 + tid; i < D; i += NTHR) A[i * LDSP + k] /= dkk;
    __syncthreads();
    for (int i = k + 1 + tid; i < D; i += NTHR) {
      const float lik = A[i * LDSP + k];
      for (int j = k + 1; j <= i; ++j) A[i * LDSP + j] -= lik * A[j * LDSP + k];
    }
    __syncthreads();
  }
}

// M = L^{-1} (lower triangular); column j solved by thread j (forward subst).
__device__ inline void trinv(const float* L, float* M, int tid) {
  if (tid < D) {
    const int j = tid;
    for (int i = 0; i < j; ++i) M[i * LDSP + j] = 0.0f;
    M[j * LDSP + j] = 1.0f / L[j * LDSP + j];
    for (int i = j + 1; i < D; ++i) {
      float acc = 0.0f;
      for (int k = j; k < i; ++k) acc += L[i * LDSP + k] * M[k * LDSP + j];
      M[i * LDSP + j] = -acc / L[i * LDSP + i];
    }
  }
  __syncthreads();
}

// C = M^T * M (64x64x64 fp32) on the matrix core: V_WMMA_F32_16X16X4_F32.
// 16 output tiles, 4 per wave, K=64 as 16 chained 16x16x4 WMMAs each.
// A(16x4): lanes 0-15 hold K0/K1, lanes 16-31 hold K2/K3 (ISA 7.12.2).
// D(16x16): VGPR r -> rows r (lanes 0-15) and r+8 (lanes 16-31).
__device__ inline void syrk_wmma(const float* M, float* C, int tid) {
  const int wave = tid >> 5;
  const int lane = tid & 31;
  const int lr = lane & 15;
  const int hi = lane >> 4;
  for (int t = wave; t < 16; t += 4) {
    const int ti = t >> 2, tj = t & 3;
    v8f acc = {};
#pragma unroll
    for (int kb = 0; kb < 16; ++kb) {
      const int k0 = kb * 4 + hi * 2;
      v2f a, b;
      a.x = M[(k0 + 0) * LDSP + ti * 16 + lr];   // A[m][k] = (M^T)[row][k] = M[k][row]
      a.y = M[(k0 + 1) * LDSP + ti * 16 + lr];
      b.x = M[(k0 + 0) * LDSP + tj * 16 + lr];   // B[k][n] = M[k][col]
      b.y = M[(k0 + 1) * LDSP + tj * 16 + lr];
      acc = __builtin_amdgcn_wmma_f32_16x16x4_f32(false, a, false, b,
                                                  (short)0, acc, false, false);
    }
#pragma unroll
    for (int r = 0; r < 8; ++r)
      C[(ti * 16 + r + hi * 8) * LDSP + tj * 16 + lr] = acc[r];
  }
  __syncthreads();
}

// y = A * x, row per thread (threads 0..63).
__device__ inline void matvec(const float* A, const float* x, float* y, int tid) {
  if (tid < D) {
    float acc = 0.0f;
    for (int j = 0; j < D; ++j) acc += A[tid * LDSP + j] * x[j];
    y[tid] = acc;
  }
  __syncthreads();
}

__global__ __launch_bounds__(NTHR)
void more_project_kernel(const float* __restrict__ tmean,
                         const float* __restrict__ tcov,
                         const float* __restrict__ omean,
                         const float* __restrict__ ocov,
                         const float* __restrict__ epss,
                         const float* __restrict__ betas,
                         float* __restrict__ out, int G) {
  __shared__ float sQ0[D * LDSP];   // precision of old covariance
  __shared__ float sQt[D * LDSP];   // precision of target covariance
  __shared__ float sW[D * LDSP];    // work: Q(eta) -> chol L -> (syrk) cov
  __shared__ float sM[D * LDSP];    // L^{-1}
  __shared__ float sOm[D], sTm[D], sq0[D], sqt[D], sLin[D], sMean[D], sTmp[D];
  __shared__ float sRed[1];

  const int tid = threadIdx.x;
  const int g = blockIdx.x;

  // ---- stage old covariance via TDM; means via plain loads ----
  if (tid < 32) tdm_load_64x64(sW, ocov + (size_t)g * D * D);
  if (tid < D) {
    sOm[tid] = omean[(size_t)g * D + tid];
    sTm[tid] = tmean[(size_t)g * D + tid];
  }
  __builtin_amdgcn_s_wait_tensorcnt(0);
  __syncthreads();

  // Q0 = inv(old_cov) via chol + tri-inv + WMMA M^T M; logdet_old from diag(L)
  chol_inplace(sW, tid);
  float dl = (tid < D) ? 2.0f * logf(sW[tid * LDSP + tid]) : 0.0f;
  const float logdet_old = block_sum(dl, sRed, tid);
  trinv(sW, sM, tid);
  syrk_wmma(sM, sQ0, tid);

  // ---- stage target covariance ----
  if (tid < 32) tdm_load_64x64(sW, tcov + (size_t)g * D * D);
  __builtin_amdgcn_s_wait_tensorcnt(0);
  __syncthreads();

  // kl0 = KL(target || old): trace + maha while sW still holds target cov
  float part = 0.0f;
  for (int idx = tid; idx < D * D; idx += NTHR) {
    const int i = idx >> 6, j = idx & 63;
    part += sQ0[i * LDSP + j] * sW[i * LDSP + j];
  }
  const float tr0 = block_sum(part, sRed, tid);
  if (tid < D) sTmp[tid] = sOm[tid] - sTm[tid];
  __syncthreads();
  matvec(sQ0, sTmp, sLin, tid);
  float md = (tid < D) ? sTmp[tid] * sLin[tid] : 0.0f;
  const float maha0 = block_sum(md, sRed, tid);

  // Qt = inv(target_cov); logdet(target_cov) from its Cholesky
  chol_inplace(sW, tid);
  dl = (tid < D) ? 2.0f * logf(sW[tid * LDSP + tid]) : 0.0f;
  const float logdet_t = block_sum(dl, sRed, tid);
  trinv(sW, sM, tid);
  syrk_wmma(sM, sQt, tid);

  const float kl0 = 0.5f * (tr0 - (float)D + maha0 + logdet_old - logdet_t);
  const float eps = epss[g];

  // natural linear params: q0 = Q0*old_mean, qt = Qt*target_mean
  matvec(sQ0, sOm, sq0, tid);
  matvec(sQt, sTm, sqt, tid);

  // ---- bisection on eta (uniform across the block; no divergence) ----
  float lo = 0.0f, hi = ETA_MAX;
  for (int it = 0; it < N_BISECT; ++it) {
    const float mid = 0.5f * (lo + hi);
    const float rinv = 1.0f / (mid + 1.0f);
    for (int idx = tid; idx < D * D; idx += NTHR) {
      const int i = idx >> 6, j = idx & 63;
      sW[i * LDSP + j] = (mid * sQ0[i * LDSP + j] + sQt[i * LDSP + j]) * rinv;
    }
    __syncthreads();
    chol_inplace(sW, tid);
    dl = (tid < D) ? 2.0f * logf(sW[tid * LDSP + tid]) : 0.0f;
    const float ldQ = block_sum(dl, sRed, tid);       // logdet(Q); logdet(cov) = -ldQ
    trinv(sW, sM, tid);
    syrk_wmma(sM, sW, tid);                           // sW := cov = M^T M (WMMA)
    part = 0.0f;
    for (int idx = tid; idx < D * D; idx += NTHR) {
      const int i = idx >> 6, j = idx & 63;
      part += sQ0[i * LDSP + j] * sW[i * LDSP + j];
    }
    const float tr = block_sum(part, sRed, tid);
    if (tid < D) sLin[tid] = (mid * sq0[tid] + sqt[tid]) * rinv;
    __syncthreads();
    matvec(sW, sLin, sMean, tid);                     // mean = cov * lin
    if (tid < D) sTmp[tid] = sOm[tid] - sMean[tid];
    __syncthreads();
    matvec(sQ0, sTmp, sLin, tid);
    md = (tid < D) ? sTmp[tid] * sLin[tid] : 0.0f;
    const float maha = block_sum(md, sRed, tid);
    const float kl = 0.5f * (tr - (float)D + maha + logdet_old + ldQ);
    const bool big = kl > eps;
    lo = big ? mid : lo;
    hi = big ? hi : mid;
  }

  // ---- final interpolation at chosen eta ----
  const float eta = (kl0 <= eps) ? 0.0f : hi;
  const float rinv = 1.0f / (eta + 1.0f);
  for (int idx = tid; idx < D * D; idx += NTHR) {
    const int i = idx >> 6, j = idx & 63;
    sW[i * LDSP + j] = (eta * sQ0[i * LDSP + j] + sQt[i * LDSP + j]) * rinv;
  }
  __syncthreads();
  chol_inplace(sW, tid);
  dl = (tid < D) ? 2.0f * logf(sW[tid * LDSP + tid]) : 0.0f;
  const float ldQ = block_sum(dl, sRed, tid);
  trinv(sW, sM, tid);
  syrk_wmma(sM, sW, tid);                             // sW := new cov
  if (tid < D) sLin[tid] = (eta * sq0[tid] + sqt[tid]) * rinv;
  __syncthreads();
  matvec(sW, sLin, sMean, tid);

  // entropy floor: logdet_new = -ldQ
  const float entropy = 0.5f * ((float)D * LOG2PIE - ldQ);
  const float beta = betas[g];
  const float scale = (entropy < beta) ? expf(2.0f * (beta - entropy) / (float)D) : 1.0f;

  float* out_means = out;
  float* out_covs = out + (size_t)G * D;
  if (tid < D) out_means[(size_t)g * D + tid] = sMean[tid];
  for (int idx = tid; idx < D * D; idx += NTHR) {
    const int i = idx >> 6, j = idx & 63;
    out_covs[(size_t)g * D * D + idx] = sW[i * LDSP + j] * scale;
  }
}

extern "C" void kernel_launch(void* const* d_in, const int* in_sizes, int n_in,
                              void* d_out, int out_size, void* d_ws, size_t ws_size,
                              hipStream_t stream) {
  (void)n_in; (void)out_size; (void)d_ws; (void)ws_size;
  const float* tmean = (const float*)d_in[0];  // q_target_means   (G, D)
  const float* tcov  = (const float*)d_in[1];  // q_target_covars  (G, D, D)
  const float* omean = (const float*)d_in[2];  // old_means        (G, D)
  const float* ocov  = (const float*)d_in[3];  // old_covs         (G, D, D)
  const float* epss  = (const float*)d_in[4];  // (G,)
  const float* betas = (const float*)d_in[5];  // (G,)
  const int G = in_sizes[4];
  more_project_kernel<<<G, NTHR, 0, stream>>>(tmean, tcov, omean, ocov, epss,
                                              betas, (float*)d_out, G);
}